// PIXLayer_42039139893965
// MI455X (gfx1250) — compile-verified
//
#include <hip/hip_runtime.h>

// out[e] = tanh(prop[idx_i[e]] @ wi1) @ wi2 + tanh(prop[idx_j[e]] @ wj1) @ wj2
// Factorized: Hi = mlp_i(prop), Hj = mlp_j(prop) over 50K nodes (WMMA f32),
// then edge phase gathers from L2-resident Hi/Hj (51.2MB << 192MB L2).

typedef __attribute__((ext_vector_type(2))) float v2f;
typedef __attribute__((ext_vector_type(4))) float v4f;
typedef __attribute__((ext_vector_type(8))) float v8f;

#define D      128
#define TILE_M 32    // nodes per block (two 16-row WMMA tiles)
#define LDSS   132   // 128 + 4 pad -> conflict-free column access across 64 banks

// CDNA5 hardware tanh (V_TANH_F32 transcendental) instead of libm's branchy
// divergent expansion.
__device__ __forceinline__ float fast_tanh(float x) {
#if __has_builtin(__builtin_amdgcn_tanhf)
    return __builtin_amdgcn_tanhf(x);
#else
    float r;
    asm volatile("v_tanh_f32 %0, %1" : "=v"(r) : "v"(x));
    return r;
#endif
}

// Two-layer MLP on a 32-node tile: Hout[node0:node0+32,:] = tanh(X @ W1) @ W2.
// Xl/Ul are LDS tiles [32 x D] with row stride LDSS. 4 waves; wave w owns
// column tiles {w, w+4} (16 cols each) and both 16-row tiles, so each B
// (weight) fragment is reused by two WMMAs.
__device__ __forceinline__ void mlp32(
    const float* __restrict__ Xl, float* __restrict__ Ul,
    const float* __restrict__ W1, const float* __restrict__ W2,
    float* __restrict__ Hout, int node0, int n_nodes,
    int wave, int lane_lo, int hs)
{
    const float* xrow0 = Xl + lane_lo * LDSS;          // rows 0..15
    const float* xrow1 = Xl + (lane_lo + 16) * LDSS;   // rows 16..31

    // ---- layer 1: U = tanh(X @ W1) ----
    #pragma unroll
    for (int tt = 0; tt < 2; ++tt) {
        const int c0 = (wave + 4 * tt) * 16;
        v8f acc0 = {0.f, 0.f, 0.f, 0.f, 0.f, 0.f, 0.f, 0.f};
        v8f acc1 = {0.f, 0.f, 0.f, 0.f, 0.f, 0.f, 0.f, 0.f};
        #pragma unroll 4
        for (int k = 0; k < D; k += 4) {
            const int ka = k + 2 * hs;          // lanes 16-31 hold K+2,K+3
            v2f a0, a1, b;
            b.x = W1[ka * D + c0 + lane_lo];    // 16 lanes -> 16 contiguous cols
            b.y = W1[(ka + 1) * D + c0 + lane_lo];
            a0.x = xrow0[ka]; a0.y = xrow0[ka + 1];
            a1.x = xrow1[ka]; a1.y = xrow1[ka + 1];
            acc0 = __builtin_amdgcn_wmma_f32_16x16x4_f32(
                false, a0, false, b, (short)0, acc0, false, false);
            acc1 = __builtin_amdgcn_wmma_f32_16x16x4_f32(
                false, a1, false, b, (short)0, acc1, false, false);
        }
        #pragma unroll
        for (int v = 0; v < 8; ++v) {
            const int m = v + 8 * hs;           // C/D layout: VGPR v -> row v (+8 hi half)
            Ul[m * LDSS + c0 + lane_lo]        = fast_tanh(acc0[v]);
            Ul[(m + 16) * LDSS + c0 + lane_lo] = fast_tanh(acc1[v]);
        }
    }
    __syncthreads();

    // ---- layer 2: H = U @ W2 ----
    const float* urow0 = Ul + lane_lo * LDSS;
    const float* urow1 = Ul + (lane_lo + 16) * LDSS;
    #pragma unroll
    for (int tt = 0; tt < 2; ++tt) {
        const int c0 = (wave + 4 * tt) * 16;
        v8f acc0 = {0.f, 0.f, 0.f, 0.f, 0.f, 0.f, 0.f, 0.f};
        v8f acc1 = {0.f, 0.f, 0.f, 0.f, 0.f, 0.f, 0.f, 0.f};
        #pragma unroll 4
        for (int k = 0; k < D; k += 4) {
            const int ka = k + 2 * hs;
            v2f a0, a1, b;
            b.x = W2[ka * D + c0 + lane_lo];
            b.y = W2[(ka + 1) * D + c0 + lane_lo];
            a0.x = urow0[ka]; a0.y = urow0[ka + 1];
            a1.x = urow1[ka]; a1.y = urow1[ka + 1];
            acc0 = __builtin_amdgcn_wmma_f32_16x16x4_f32(
                false, a0, false, b, (short)0, acc0, false, false);
            acc1 = __builtin_amdgcn_wmma_f32_16x16x4_f32(
                false, a1, false, b, (short)0, acc1, false, false);
        }
        #pragma unroll
        for (int v = 0; v < 8; ++v) {
            const int m = v + 8 * hs;
            const int r0 = node0 + m;
            const int r1 = node0 + m + 16;
            if (r0 < n_nodes) Hout[(size_t)r0 * D + c0 + lane_lo] = acc0[v];
            if (r1 < n_nodes) Hout[(size_t)r1 * D + c0 + lane_lo] = acc1[v];
        }
    }
}

__global__ __launch_bounds__(128) void node_mlp_kernel(
    const float* __restrict__ prop,
    const float* __restrict__ wi1, const float* __restrict__ wi2,
    const float* __restrict__ wj1, const float* __restrict__ wj2,
    float* __restrict__ hi, float* __restrict__ hj, int n_nodes)
{
    __shared__ __align__(16) float X[TILE_M * LDSS];
    __shared__ __align__(16) float U[TILE_M * LDSS];

    const int t       = threadIdx.x;
    const int wave    = t >> 5;
    const int lane    = t & 31;
    const int lane_lo = lane & 15;
    const int hs      = lane >> 4;
    const int node0   = blockIdx.x * TILE_M;

    // Stage X tile (32 x 128 f32) into LDS, 128B-coalesced float4 loads.
    #pragma unroll
    for (int i = 0; i < 8; ++i) {
        const int gid = t + i * 128;       // 0..1023 float4 slots
        const int row = gid >> 5;          // 0..31
        const int c4  = gid & 31;          // 0..31
        int srow = node0 + row;
        if (srow >= n_nodes) srow = n_nodes - 1;   // clamp tail (stores guarded later)
        v4f v = *(const v4f*)(prop + (size_t)srow * D + c4 * 4);
        *(v4f*)(X + row * LDSS + c4 * 4) = v;
    }
    __syncthreads();

    mlp32(X, U, wi1, wi2, hi, node0, n_nodes, wave, lane_lo, hs);
    __syncthreads();
    mlp32(X, U, wj1, wj2, hj, node0, n_nodes, wave, lane_lo, hs);
}

// out[e,:] = hi[idx_i[e],:] + hj[idx_j[e],:]
// One wave per edge row: 32 lanes x float4 = 128 floats = 512B contiguous.
// hi/hj gathers hit L2 (51.2MB resident); out streamed non-temporally so it
// doesn't evict them.
__global__ __launch_bounds__(256) void edge_add_kernel(
    const int* __restrict__ idx_i, const int* __restrict__ idx_j,
    const float* __restrict__ hi, const float* __restrict__ hj,
    float* __restrict__ out, int n_edges)
{
    const int lane = threadIdx.x & 31;
    const int wid  = (int)((blockIdx.x * blockDim.x + threadIdx.x) >> 5);
    const int nw   = (int)((gridDim.x * blockDim.x) >> 5);
    for (int e = wid; e < n_edges; e += nw) {
        const int ni = idx_i[e];
        const int nj = idx_j[e];
        v4f a = *(const v4f*)(hi + (size_t)ni * D + lane * 4);
        v4f b = *(const v4f*)(hj + (size_t)nj * D + lane * 4);
        v4f r = a + b;
        __builtin_nontemporal_store(r, (v4f*)(out + (size_t)e * D) + lane);
    }
}

extern "C" void kernel_launch(void* const* d_in, const int* in_sizes, int n_in,
                              void* d_out, int out_size, void* d_ws, size_t ws_size,
                              hipStream_t stream) {
    const float* prop = (const float*)d_in[0];
    const int*   idx_i = (const int*)d_in[1];
    const int*   idx_j = (const int*)d_in[2];
    const float* wi1  = (const float*)d_in[3];
    const float* wi2  = (const float*)d_in[4];
    const float* wj1  = (const float*)d_in[5];
    const float* wj2  = (const float*)d_in[6];
    float* out = (float*)d_out;

    const int n_nodes = in_sizes[0] / D;   // 50000
    const int n_edges = in_sizes[1];       // 800000

    // Workspace: Hi then Hj, each n_nodes x 128 f32 (2 x 25.6 MB).
    float* hi = (float*)d_ws;
    float* hj = hi + (size_t)n_nodes * D;

    // Phase 1: per-node MLPs via f32 WMMA. 32 nodes per block, 4 waves/block.
    const int g1 = (n_nodes + TILE_M - 1) / TILE_M;
    node_mlp_kernel<<<g1, 128, 0, stream>>>(prop, wi1, wi2, wj1, wj2, hi, hj, n_nodes);

    // Phase 2: edge gather + add. ~4 edges per wave, 8 waves per block.
    int blocks2 = (n_edges + 31) / 32;     // total waves ~= edges/4
    if (blocks2 < 1) blocks2 = 1;
    edge_add_kernel<<<blocks2, 256, 0, stream>>>(idx_i, idx_j, hi, hj, out, n_edges);
}